// GNNLoRA_40578851013018
// MI455X (gfx1250) — compile-verified
//
#include <hip/hip_runtime.h>

// ---------------------------------------------------------------------------
// GAT + LoRA pipeline for MI455X (gfx1250, wave32).
//  - Dense node transforms: fp32 WMMA (V_WMMA_F32_16X16X4_F32), LDS-staged
//    operands via async global->LDS DMA (GLOBAL_LOAD_ASYNC_TO_LDS_B128,
//    ASYNCcnt), 128-row x Nc block tiles, NT-way A-fragment reuse.
//  - LoRA low-rank factors pre-multiplied into one dense weight per layer.
//  - Segment softmax: 3 edge passes with L2-resident f32 atomics
//    (features + accumulators are 51MB << 192MB L2).
// ---------------------------------------------------------------------------

typedef float v2f __attribute__((ext_vector_type(2)));
typedef float v8f __attribute__((ext_vector_type(8)));

#define NNODES   100000
#define NEDGES   1600000
#define D_IN     128
#define D_OUT    64
#define D_HID    128
#define EA       (NEDGES + NNODES)   // edges incl. self loops
#define NEG_SLOPE 0.2f

static __device__ __forceinline__ void atomicMaxF(float* addr, float v) {
    // sign-aware monotonic-bits trick; m initialized to -inf
    if (v >= 0.0f) atomicMax((int*)addr, __float_as_int(v));
    else           atomicMin((unsigned int*)addr, __float_as_uint(v));
}

// async global -> LDS copy of one 16B chunk per lane (gfx1250 TDM-adjacent path)
static __device__ __forceinline__ void async_g2l_b128(unsigned lds_byte_addr,
                                                      unsigned global_byte_off,
                                                      unsigned long long base) {
    asm volatile("global_load_async_to_lds_b128 %0, %1, %2"
                 :: "v"(lds_byte_addr), "v"(global_byte_off), "s"(base)
                 : "memory");
}
static __device__ __forceinline__ void wait_async0() {
    asm volatile("s_wait_asynccnt 0x0" ::: "memory");
}

// ---------------- utility fills ----------------
__global__ void k_fill(float* __restrict__ p, float v, long long n) {
    long long i = (long long)blockIdx.x * blockDim.x + threadIdx.x;
    if (i < n) p[i] = v;
}

// ---------------- edge list (int64 -> int32, append self loops) ------------
__global__ void k_build_edges(const long long* __restrict__ ei,
                              int* __restrict__ src, int* __restrict__ dst) {
    int i = blockIdx.x * blockDim.x + threadIdx.x;
    if (i >= EA) return;
    if (i < NEDGES) {
        src[i] = (int)ei[i];
        dst[i] = (int)ei[(long long)NEDGES + i];
    } else {
        src[i] = i - NEDGES;
        dst[i] = i - NEDGES;
    }
}

// ---------------- fp32 WMMA GEMM with LDS staging ---------------------------
// C[M, NT*16] = A[M,K] * B[K, NT*16].
// Block = 256 threads (8 waves) computes a 128 x Nc tile.
// A block (128 x K) and full B panel (K x Nc) staged in padded LDS via
// async global->LDS b128; wave w computes rows [w*16, w*16+16) x all Nc.
template<int K, int NT>
__global__ void k_gemm_wmma_lds(const float* __restrict__ A,
                                const float* __restrict__ B,
                                float* __restrict__ C, int M) {
    constexpr int Nc = NT * 16;
    constexpr int AP = K + 4;    // padded A row stride (floats) -> bank spread
    constexpr int BP = Nc + 4;   // padded B row stride (floats)

    extern __shared__ __align__(16) float lds[];
    float* lsA = lds;                 // 128 x AP
    float* lsB = lds + 128 * AP;      // K x BP

    const int tid  = threadIdx.x;
    const int lane = tid & 31;
    const int wave = tid >> 5;
    const int baseRow = blockIdx.x * 128;

    const unsigned long long abase = (unsigned long long)A;
    const unsigned long long bbase = (unsigned long long)B;

    // ---- stage A block: 128 x K floats, 16B chunks, rows clamped at tail ----
    constexpr int ACH = (128 * K) / 4;          // 16B chunks
#pragma unroll
    for (int c = tid; c < ACH; c += 256) {
        int lr  = c / (K / 4);
        int col = (c % (K / 4)) * 4;
        int gr  = baseRow + lr; if (gr > M - 1) gr = M - 1;
        unsigned goff = (unsigned)(((long long)gr * K + col) * 4);
        unsigned loff = (unsigned)(unsigned long long)(lsA + lr * AP + col);
        async_g2l_b128(loff, goff, abase);
    }
    // ---- stage B panel: K x Nc floats ----
    constexpr int BCH = (K * Nc) / 4;
#pragma unroll
    for (int c = tid; c < BCH; c += 256) {
        int r   = c / (Nc / 4);
        int col = (c % (Nc / 4)) * 4;
        unsigned goff = (unsigned)((r * Nc + col) * 4);
        unsigned loff = (unsigned)(unsigned long long)(lsB + r * BP + col);
        async_g2l_b128(loff, goff, bbase);
    }
    wait_async0();
    __syncthreads();

    if (baseRow + wave * 16 < M) {
        const int half = lane >> 4;   // 0: K pair {0,1}, 1: K pair {2,3}
        const int l16  = lane & 15;
        const float* aRow = lsA + (wave * 16 + l16) * AP + (half << 1);
        const float* bCol = lsB + l16;

        v8f zero = {};
        v8f acc[NT];
#pragma unroll
        for (int t = 0; t < NT; ++t) acc[t] = zero;

#pragma unroll 4
        for (int k = 0; k < K; k += 4) {
            v2f a;
            a.x = aRow[k];
            a.y = aRow[k + 1];
            const float* b0 = bCol + (k + (half << 1)) * BP;
#pragma unroll
            for (int t = 0; t < NT; ++t) {
                v2f b;
                b.x = b0[t * 16];
                b.y = b0[t * 16 + BP];
                acc[t] = __builtin_amdgcn_wmma_f32_16x16x4_f32(
                    false, a, false, b, (short)0, acc[t], false, false);
            }
        }
        // C/D layout: VGPR i -> row (+8*half + i), col l16
#pragma unroll
        for (int t = 0; t < NT; ++t) {
            float* cp = C + (size_t)(baseRow + wave * 16 + (half << 3)) * Nc
                          + t * 16 + l16;
#pragma unroll
            for (int i = 0; i < 8; ++i) cp[(size_t)i * Nc] = acc[t][i];
        }
    }
}

template<int K, int NT>
static void gemm(const float* A, const float* B, float* C, int M, hipStream_t st) {
    int blocks = (M + 127) / 128;
    size_t smem = (size_t)(128 * (K + 4) + K * (NT * 16 + 4)) * sizeof(float);
    k_gemm_wmma_lds<K, NT><<<blocks, 256, smem, st>>>(A, B, C, M);
}

// ---------------- per-node attention logits: a_s, a_d -----------------------
__global__ void k_rowdots(const float* __restrict__ xt,
                          const float* __restrict__ att_s,
                          const float* __restrict__ att_d,
                          float* __restrict__ as_, float* __restrict__ ad_,
                          int n, int C) {
    int lane = threadIdx.x & 31;
    int w = blockIdx.x * (blockDim.x >> 5) + (threadIdx.x >> 5);
    if (w >= n) return;
    const float* row = xt + (size_t)w * C;
    float s = 0.f, d = 0.f;
    for (int c = lane; c < C; c += 32) {
        float v = row[c];
        s += v * att_s[c];
        d += v * att_d[c];
    }
#pragma unroll
    for (int off = 16; off > 0; off >>= 1) {
        s += __shfl_down(s, off, 32);
        d += __shfl_down(d, off, 32);
    }
    if (lane == 0) { as_[w] = s; ad_[w] = d; }
}

// ---------------- edge pass 1: leaky-relu logit + segment max ---------------
__global__ void k_edge_max(const int* __restrict__ src, const int* __restrict__ dst,
                           const float* __restrict__ as_, const float* __restrict__ ad_,
                           float* __restrict__ e, float* __restrict__ m) {
    int i = blockIdx.x * blockDim.x + threadIdx.x;
    if (i >= EA) return;
    float v = as_[src[i]] + ad_[dst[i]];
    v = (v > 0.f) ? v : NEG_SLOPE * v;
    e[i] = v;
    atomicMaxF(&m[dst[i]], v);
}

// ---------------- edge pass 2: exp + segment sum ----------------------------
__global__ void k_edge_expsum(const int* __restrict__ dst,
                              float* __restrict__ e,
                              const float* __restrict__ m,
                              float* __restrict__ s) {
    int i = blockIdx.x * blockDim.x + threadIdx.x;
    if (i >= EA) return;
    int d = dst[i];
    float ex = __expf(e[i] - m[d]);
    e[i] = ex;
    atomicAdd(&s[d], ex);
}

// ---------------- edge pass 3: weighted gather/scatter-add ------------------
// one wave per edge; lane owns C/32 contiguous channels (float4/float2 gather,
// unrolled global_atomic_add_f32 scatter, both L2-resident)
template<int C>
__global__ void k_edge_scatter(const int* __restrict__ src, const int* __restrict__ dst,
                               const float* __restrict__ e, const float* __restrict__ s,
                               const float* __restrict__ xt, float* __restrict__ out) {
    constexpr int V = C / 32;    // floats per lane
    int lane = threadIdx.x & 31;
    int w = blockIdx.x * (blockDim.x >> 5) + (threadIdx.x >> 5);
    if (w >= EA) return;
    int sv = src[w], dv = dst[w];
    float att = e[w] / (s[dv] + 1e-16f);
    const float* xr = xt + (size_t)sv * C + lane * V;
    float* orow     = out + (size_t)dv * C + lane * V;
    float buf[V];
#pragma unroll
    for (int i = 0; i < V; ++i) buf[i] = xr[i];
#pragma unroll
    for (int i = 0; i < V; ++i) atomicAdd(&orow[i], buf[i] * att);
}

// ---------------- bias (+ optional relu), in place --------------------------
__global__ void k_bias_act(float* __restrict__ h, const float* __restrict__ bias,
                           long long total, int Cmask, int relu) {
    long long i = (long long)blockIdx.x * blockDim.x + threadIdx.x;
    if (i >= total) return;
    float v = h[i] + bias[(int)i & Cmask];   // C is power of two (128/64)
    if (relu) v = fmaxf(v, 0.f);
    h[i] = v;
}

// ---------------- final: out0 = xb + xl -------------------------------------
__global__ void k_add(float* __restrict__ o, const float* __restrict__ a,
                      const float* __restrict__ b, long long n) {
    long long i = (long long)blockIdx.x * blockDim.x + threadIdx.x;
    if (i < n) o[i] = a[i] + b[i];
}

// ---------------------------------------------------------------------------
static inline int cdiv(long long a, int b) { return (int)((a + b - 1) / b); }

// one full GAT conv on already-transformed features xt[n,C] -> out[n,C]
template<int C>
static void run_conv(const float* xt, const float* att_s, const float* att_d,
                     const float* bias, float* out, int relu,
                     const int* src, const int* dst,
                     float* a_s, float* a_d, float* m, float* s, float* e,
                     hipStream_t st) {
    const int n = NNODES;
    k_rowdots<<<cdiv(n, 8), 256, 0, st>>>(xt, att_s, att_d, a_s, a_d, n, C);
    k_fill<<<cdiv(n, 256), 256, 0, st>>>(m, -__builtin_huge_valf(), n);
    k_fill<<<cdiv(n, 256), 256, 0, st>>>(s, 0.f, n);
    k_fill<<<cdiv((long long)n * C, 256), 256, 0, st>>>(out, 0.f, (long long)n * C);
    k_edge_max<<<cdiv(EA, 256), 256, 0, st>>>(src, dst, a_s, a_d, e, m);
    k_edge_expsum<<<cdiv(EA, 256), 256, 0, st>>>(dst, e, m, s);
    k_edge_scatter<C><<<cdiv(EA, 8), 256, 0, st>>>(src, dst, e, s, xt, out);
    k_bias_act<<<cdiv((long long)n * C, 256), 256, 0, st>>>(out, bias,
                                                            (long long)n * C, C - 1, relu);
}

extern "C" void kernel_launch(void* const* d_in, const int* in_sizes, int n_in,
                              void* d_out, int out_size, void* d_ws, size_t ws_size,
                              hipStream_t stream) {
    // inputs in setup_inputs() order
    const float*     x   = (const float*)d_in[0];
    const long long* ei  = (const long long*)d_in[1];   // jnp.int64 per reference
    const float* W0  = (const float*)d_in[2];
    const float* as0 = (const float*)d_in[3];
    const float* ad0 = (const float*)d_in[4];
    const float* b0  = (const float*)d_in[5];
    const float* W1  = (const float*)d_in[6];
    const float* as1 = (const float*)d_in[7];
    const float* ad1 = (const float*)d_in[8];
    const float* b1  = (const float*)d_in[9];
    const float* A0  = (const float*)d_in[10];
    const float* B0  = (const float*)d_in[11];
    const float* las0 = (const float*)d_in[12];
    const float* lad0 = (const float*)d_in[13];
    const float* lb0  = (const float*)d_in[14];
    const float* A1  = (const float*)d_in[15];
    const float* B1  = (const float*)d_in[16];
    const float* las1 = (const float*)d_in[17];
    const float* lad1 = (const float*)d_in[18];
    const float* lb1  = (const float*)d_in[19];

    float* out_sum = (float*)d_out;                                  // xb + xl
    float* out_xb  = (float*)d_out + (long long)NNODES * D_OUT;      // xb
    float* out_xl  = (float*)d_out + 2LL * NNODES * D_OUT;           // xl

    // ---- workspace carve-up ----
    char* w = (char*)d_ws;
    size_t off = 0;
    auto take = [&](size_t bytes) {
        void* p = w + off;
        off = (off + bytes + 255) & ~(size_t)255;
        return p;
    };
    int*   srcE = (int*)  take(sizeof(int)   * EA);
    int*   dstE = (int*)  take(sizeof(int)   * EA);
    float* ebuf = (float*)take(sizeof(float) * EA);
    float* a_s  = (float*)take(sizeof(float) * NNODES);
    float* a_d  = (float*)take(sizeof(float) * NNODES);
    float* m    = (float*)take(sizeof(float) * NNODES);
    float* ssum = (float*)take(sizeof(float) * NNODES);
    float* xt   = (float*)take(sizeof(float) * (size_t)NNODES * D_HID); // 51.2 MB
    float* h    = (float*)take(sizeof(float) * (size_t)NNODES * D_HID); // 51.2 MB
    float* Wl0  = (float*)take(sizeof(float) * D_IN  * D_HID);
    float* Wl1  = (float*)take(sizeof(float) * D_HID * D_OUT);
    (void)ws_size; (void)in_sizes; (void)n_in; (void)out_size;

    // ---- edge list with self loops, int32 ----
    k_build_edges<<<cdiv(EA, 256), 256, 0, stream>>>(ei, srcE, dstE);

    // ---- merge LoRA factors: Wl = A @ B (tiny WMMA GEMMs) ----
    gemm<32, 8>(A0, B0, Wl0, D_IN,  stream);   // 128x32 @ 32x128
    gemm<32, 4>(A1, B1, Wl1, D_HID, stream);   // 128x32 @ 32x64

    // ---- base path ----
    gemm<128, 8>(x, W0, xt, NNODES, stream);                          // xt = x@W0
    run_conv<D_HID>(xt, as0, ad0, b0, h, /*relu=*/1,
                    srcE, dstE, a_s, a_d, m, ssum, ebuf, stream);     // h = relu(gat0)
    gemm<128, 4>(h, W1, xt, NNODES, stream);                          // xt = h@W1
    run_conv<D_OUT>(xt, as1, ad1, b1, out_xb, /*relu=*/0,
                    srcE, dstE, a_s, a_d, m, ssum, ebuf, stream);     // xb

    // ---- LoRA path ----
    gemm<128, 8>(x, Wl0, xt, NNODES, stream);                         // xt = x@(A0B0)
    run_conv<D_HID>(xt, las0, lad0, lb0, h, /*relu=*/1,
                    srcE, dstE, a_s, a_d, m, ssum, ebuf, stream);     // h = relu(gatL0)
    gemm<128, 4>(h, Wl1, xt, NNODES, stream);                         // xt = h@(A1B1)
    run_conv<D_OUT>(xt, las1, lad1, lb1, out_xl, /*relu=*/0,
                    srcE, dstE, a_s, a_d, m, ssum, ebuf, stream);     // xl

    // ---- out0 = xb + xl ----
    k_add<<<cdiv((long long)NNODES * D_OUT, 256), 256, 0, stream>>>(
        out_sum, out_xb, out_xl, (long long)NNODES * D_OUT);
}